// AttnEdgeConv_18227841204609
// MI455X (gfx1250) — compile-verified
//
#include <hip/hip_runtime.h>
#include <hip/hip_bf16.h>
#include <math.h>

typedef float v8f __attribute__((ext_vector_type(8)));
typedef float v2f __attribute__((ext_vector_type(2)));
typedef int   v4i __attribute__((vector_size(16)));   // 4 x i32, matches builtin

#define AS1 __attribute__((address_space(1)))
#define AS3 __attribute__((address_space(3)))

#define NGRAPH 16
#define NNODE  2048
#define CIN    16
#define OOUT   64
#define KNN2   16
#define TOTALN (NGRAPH * NNODE)   // 32768
#define NEDGE  (TOTALN * KNN2)    // 524288
#define HPAD   68                 // 64 + 4 pad: conflict-free column access
#define CHUNK  512
#define EPSBN  1e-5f

// ---------------------------------------------------------------------------
// gfx1250 async global->LDS copy (ASYNCcnt path) with compile-safe fallback
// ---------------------------------------------------------------------------
__device__ __forceinline__ void async_copy_b128(const float* g, float* l) {
#if __has_builtin(__builtin_amdgcn_global_load_async_to_lds_b128)
    __builtin_amdgcn_global_load_async_to_lds_b128(
        (AS1 v4i*)const_cast<float*>(g), (AS3 v4i*)l, 0, 0);
#else
    float4 v = *(const float4*)g;
    *(float4*)l = v;
#endif
}

__device__ __forceinline__ void wait_async0() {
#if __has_builtin(__builtin_amdgcn_s_wait_asynccnt)
    __builtin_amdgcn_s_wait_asynccnt(0);
#else
    asm volatile("s_wait_asynccnt 0x0" ::: "memory");
#endif
}

// stage W1 [32x64] into LDS: 512 x b128 transfers across 256 threads
__device__ __forceinline__ void stage_w1(const float* __restrict__ W1,
                                         float* __restrict__ w1s, int tid) {
#pragma unroll
    for (int i = 0; i < 2; ++i) {
        int e = (tid + i * 256) * 4;          // float index, 16B aligned
        async_copy_b128(W1 + e, &w1s[e]);
    }
    wait_async0();
}

// ---------------------------------------------------------------------------
// K1: per-node squared norms
// ---------------------------------------------------------------------------
__global__ __launch_bounds__(256) void sqnorm_kernel(const float* __restrict__ x,
                                                     float* __restrict__ sq) {
    int i = blockIdx.x * blockDim.x + threadIdx.x;
    if (i < TOTALN) {
        const float4* xr = (const float4*)(x + (size_t)i * CIN);
        float s = 0.f;
#pragma unroll
        for (int j = 0; j < 4; ++j) {
            float4 v = xr[j];
            s += v.x * v.x + v.y * v.y + v.z * v.z + v.w * v.w;
        }
        sq[i] = s;
    }
}

// ---------------------------------------------------------------------------
// top-16 register insertion (fully unrolled -> stays in VGPRs)
// ---------------------------------------------------------------------------
__device__ __forceinline__ void topk_insert(float (&td)[16], int (&ti)[16],
                                            float d, int di) {
    if (d >= td[15]) return;
#pragma unroll
    for (int i = 15; i >= 0; --i) {
        float cur = td[i];
        if (cur > d) {
            float pd = (i > 0) ? td[i - 1] : d;
            int   pi = (i > 0) ? ti[i - 1] : di;
            bool shift = (i > 0) && (pd > d);
            td[i] = shift ? pd : d;
            ti[i] = shift ? pi : di;
        }
    }
}

// ---------------------------------------------------------------------------
// K2: dynamic kNN per graph.  One block = (graph, 16-row tile).
// Gram tiles via V_WMMA_F32_16X16X4_F32; distances chunked through LDS;
// per-thread top-16 over a column sub-range; 8-way merge per row.
// ---------------------------------------------------------------------------
__global__ __launch_bounds__(128) void knn_kernel(const float* __restrict__ x,
                                                  const float* __restrict__ sq,
                                                  int* __restrict__ idx_out) {
    __shared__ float distb[16][CHUNK];     // 32 KB
    __shared__ float md[16][8][16];        // 8 KB
    __shared__ int   mi[16][8][16];        // 8 KB
    __shared__ int   heads[16][8];

    const int g        = blockIdx.x / (NNODE / 16);
    const int rb       = blockIdx.x % (NNODE / 16);
    const int gbase    = g * NNODE;
    const int growbase = gbase + rb * 16;

    const int lane  = threadIdx.x & 31;
    const int wave  = threadIdx.x >> 5;
    const int m_a   = lane & 15;            // row (A) / col (B) within tile
    const int off   = (lane >> 4) << 1;     // K sub-offset: 0 or 2
    const int mhalf = (lane >> 4) * 8;      // C/D tile: M base for this half

    // A-matrix registers: the 16-row tile of x (fixed for whole block)
    v2f a[4];
    {
        const float* xrow = x + (size_t)(growbase + m_a) * CIN;
#pragma unroll
        for (int j = 0; j < 4; ++j) {
            a[j].x = xrow[4 * j + off];
            a[j].y = xrow[4 * j + off + 1];
        }
    }
    float sqr[8];
#pragma unroll
    for (int r = 0; r < 8; ++r) sqr[r] = sq[growbase + mhalf + r];

    // per-thread running top-16 (row = tid/8, scans 1/8 of the columns)
    float td[16]; int ti[16];
#pragma unroll
    for (int i = 0; i < 16; ++i) { td[i] = 3.0e38f; ti[i] = -1; }
    const int row = threadIdx.x >> 3;   // 0..15
    const int sub = threadIdx.x & 7;    // 0..7

    for (int ch = 0; ch < NNODE / CHUNK; ++ch) {
        const int chbase = ch * CHUNK;
        // waves compute interleaved 16-col tiles of the distance chunk
        for (int t = wave; t < CHUNK / 16; t += 4) {
            const int colbase = chbase + t * 16;
            const float* xcol = x + (size_t)(gbase + colbase + m_a) * CIN;
            // prefetch next tile this wave will process (L2-speculative)
            __builtin_prefetch(xcol + 4 * 16 * CIN, 0, 1);
            v8f acc = {0.f, 0.f, 0.f, 0.f, 0.f, 0.f, 0.f, 0.f};
#pragma unroll
            for (int j = 0; j < 4; ++j) {
                v2f b;
                b.x = xcol[4 * j + off];
                b.y = xcol[4 * j + off + 1];
                acc = __builtin_amdgcn_wmma_f32_16x16x4_f32(
                    false, a[j], false, b, (short)0, acc, false, false);
            }
            const float sqc = sq[gbase + colbase + m_a];
#pragma unroll
            for (int r = 0; r < 8; ++r) {
                distb[mhalf + r][t * 16 + m_a] = sqr[r] + sqc - 2.0f * acc[r];
            }
        }
        __syncthreads();
        // selection scan over this chunk: float4 LDS reads (4x fewer DS ops)
        const int base = sub * 64;
        const float4* rowp = (const float4*)&distb[row][base];
        const int cb0 = gbase + chbase + base;
        for (int q = 0; q < 16; ++q) {
            float4 d4 = rowp[q];
            const int ci = cb0 + q * 4;
            topk_insert(td, ti, d4.x, ci + 0);
            topk_insert(td, ti, d4.y, ci + 1);
            topk_insert(td, ti, d4.z, ci + 2);
            topk_insert(td, ti, d4.w, ci + 3);
        }
        __syncthreads();
    }

    // dump sorted locals, then 8-way merge per row (threads 0..15)
#pragma unroll
    for (int i = 0; i < 16; ++i) { md[row][sub][i] = td[i]; mi[row][sub][i] = ti[i]; }
    heads[row][sub] = 0;
    __syncthreads();

    if (threadIdx.x < 16) {
        const int r = threadIdx.x;
        for (int k = 0; k < KNN2; ++k) {
            float best = 3.3e38f; int bs = 0;
#pragma unroll
            for (int s = 0; s < 8; ++s) {
                int h = heads[r][s];
                float v = (h < 16) ? md[r][s][h] : 3.3e38f;
                if (v < best) { best = v; bs = s; }
            }
            int h = heads[r][bs];
            idx_out[(size_t)(growbase + r) * KNN2 + k] = mi[r][bs][h];
            heads[r][bs] = h + 1;
        }
    }
}

// ---------------------------------------------------------------------------
// Edge MLP: one wave computes one node's 16 edges (M-tile) against W1 [32x64]
// using 8 K-steps x 4 N-tiles of V_WMMA_F32_16X16X4_F32.  Raw h (+bias) is
// written to this wave's LDS tile hb[16][HPAD].
// ---------------------------------------------------------------------------
__device__ __forceinline__ void wave_compute_h(int node, int lane,
                                               const float* __restrict__ x,
                                               const int* __restrict__ nbr,
                                               const float* __restrict__ w1s,  // LDS [32*64]
                                               const float* __restrict__ b1,
                                               float* __restrict__ hb) {       // LDS 16*HPAD
    const int m     = lane & 15;
    const int off   = (lane >> 4) << 1;
    const int mhalf = (lane >> 4) * 8;

    const float* xn = x + (size_t)node * CIN;
    const int nb    = nbr[(size_t)node * KNN2 + m];
    const float* xm = x + (size_t)nb * CIN;

    // A: e[m][0..31] = [ x_i , x_j - x_i ]  (this lane's 16-float slice)
    v2f a[8];
#pragma unroll
    for (int j = 0; j < 4; ++j) {
        a[j].x = xn[4 * j + off];
        a[j].y = xn[4 * j + off + 1];
    }
#pragma unroll
    for (int j = 0; j < 4; ++j) {
        a[4 + j].x = xm[4 * j + off]     - a[j].x;
        a[4 + j].y = xm[4 * j + off + 1] - a[j].y;
    }

    v8f acc[4];
#pragma unroll
    for (int t = 0; t < 4; ++t) acc[t] = (v8f){0.f,0.f,0.f,0.f,0.f,0.f,0.f,0.f};

#pragma unroll
    for (int j = 0; j < 8; ++j) {
#pragma unroll
        for (int t = 0; t < 4; ++t) {
            v2f b;
            b.x = w1s[(4 * j + off)     * OOUT + t * 16 + m];
            b.y = w1s[(4 * j + off + 1) * OOUT + t * 16 + m];
            acc[t] = __builtin_amdgcn_wmma_f32_16x16x4_f32(
                false, a[j], false, b, (short)0, acc[t], false, false);
        }
    }
#pragma unroll
    for (int t = 0; t < 4; ++t) {
        const int c = t * 16 + m;
        const float bias = b1[c];
#pragma unroll
        for (int r = 0; r < 8; ++r)
            hb[(mhalf + r) * HPAD + c] = acc[t][r] + bias;
    }
}

// ---------------------------------------------------------------------------
// K3: pass 1 — raw h, per-block deterministic BN partial sums
// block = 256 threads = 8 waves = 8 nodes
// ---------------------------------------------------------------------------
__global__ __launch_bounds__(256) void mlp_stats_kernel(const float* __restrict__ x,
                                                        const int* __restrict__ nbr,
                                                        const float* __restrict__ W1,
                                                        const float* __restrict__ b1,
                                                        float* __restrict__ hpart) {
    __shared__ float w1s[2 * CIN * OOUT];
    __shared__ float hb[8][16 * HPAD];
    __shared__ float wsum[8][OOUT];
    __shared__ float wsum2[8][OOUT];

    const int tid = threadIdx.x;
    stage_w1(W1, w1s, tid);
    __syncthreads();

    const int wave = tid >> 5, lane = tid & 31;
    const int node = blockIdx.x * 8 + wave;
    wave_compute_h(node, lane, x, nbr, w1s, b1, &hb[wave][0]);
    __syncthreads();

    const int c0 = lane, c1 = lane + 32;
    float s0 = 0.f, q0 = 0.f, s1 = 0.f, q1 = 0.f;
#pragma unroll
    for (int k = 0; k < 16; ++k) {
        float v = hb[wave][k * HPAD + c0]; s0 += v; q0 += v * v;
        float w = hb[wave][k * HPAD + c1]; s1 += w; q1 += w * w;
    }
    wsum[wave][c0] = s0; wsum2[wave][c0] = q0;
    wsum[wave][c1] = s1; wsum2[wave][c1] = q1;
    __syncthreads();

    if (tid < 64) {
        float s = 0.f;
#pragma unroll
        for (int w = 0; w < 8; ++w) s += wsum[w][tid];
        hpart[(size_t)blockIdx.x * 128 + tid] = s;
    } else if (tid < 128) {
        const int c = tid - 64;
        float s = 0.f;
#pragma unroll
        for (int w = 0; w < 8; ++w) s += wsum2[w][c];
        hpart[(size_t)blockIdx.x * 128 + tid] = s;
    }
}

// K3b: deterministic reduction of BN partials
__global__ void reduce_hstats_kernel(const float* __restrict__ hpart,
                                     float* __restrict__ hstats) {
    const int c = threadIdx.x;   // 0..127
    float s = 0.f;
    for (int b = 0; b < TOTALN / 8; ++b) s += hpart[(size_t)b * 128 + c];
    hstats[c] = s;
}

// BN + SiLU in place on this wave's hb tile (lane handles channels lane, lane+32)
__device__ __forceinline__ void apply_bn_silu(float* __restrict__ hb, int lane,
                                              const float* __restrict__ hstats,
                                              const float* __restrict__ g1,
                                              const float* __restrict__ be1) {
    const float invn = 1.0f / (float)NEDGE;
#pragma unroll
    for (int half = 0; half < 2; ++half) {
        const int c = lane + 32 * half;
        const float mu = hstats[c] * invn;
        const float var = hstats[64 + c] * invn - mu * mu;
        const float rstd = rsqrtf(var + EPSBN);
        const float ga = g1[c], bb = be1[c];
#pragma unroll
        for (int k = 0; k < 16; ++k) {
            float v = hb[k * HPAD + c];
            v = (v - mu) * rstd * ga + bb;
            v = v / (1.0f + __expf(-v));   // SiLU
            hb[k * HPAD + c] = v;
        }
    }
}

// ---------------------------------------------------------------------------
// K5: pass 2 — recompute h, BN+SiLU, gate gt = h@Wg + bg, gate stat partials
// ---------------------------------------------------------------------------
__global__ __launch_bounds__(256) void gate_kernel(const float* __restrict__ x,
                                                   const int* __restrict__ nbr,
                                                   const float* __restrict__ W1,
                                                   const float* __restrict__ b1,
                                                   const float* __restrict__ g1,
                                                   const float* __restrict__ be1,
                                                   const float* __restrict__ hstats,
                                                   const float* __restrict__ Wg,
                                                   const float* __restrict__ bg,
                                                   float* __restrict__ gt_out,
                                                   float* __restrict__ gpart) {
    __shared__ float w1s[2 * CIN * OOUT];
    __shared__ float hb[8][16 * HPAD];
    __shared__ float wg[8], wg2[8];

    const int tid = threadIdx.x;
    stage_w1(W1, w1s, tid);
    __syncthreads();

    const int wave = tid >> 5, lane = tid & 31;
    const int node = blockIdx.x * 8 + wave;
    wave_compute_h(node, lane, x, nbr, w1s, b1, &hb[wave][0]);
    __syncthreads();
    apply_bn_silu(&hb[wave][0], lane, hstats, g1, be1);
    __syncthreads();

    if (lane < 16) {
        float acc = bg[0];
        for (int c = 0; c < OOUT; ++c) acc += hb[wave][lane * HPAD + c] * Wg[c];
        gt_out[(size_t)node * KNN2 + lane] = acc;
        float s = acc, q = acc * acc;
#pragma unroll
        for (int mask = 1; mask < 16; mask <<= 1) {
            s += __shfl_xor(s, mask);
            q += __shfl_xor(q, mask);
        }
        if (lane == 0) { wg[wave] = s; wg2[wave] = q; }
    }
    __syncthreads();
    if (tid == 0) {
        float s = 0.f, q = 0.f;
#pragma unroll
        for (int w = 0; w < 8; ++w) { s += wg[w]; q += wg2[w]; }
        gpart[(size_t)blockIdx.x * 2 + 0] = s;
        gpart[(size_t)blockIdx.x * 2 + 1] = q;
    }
}

// K5b: deterministic reduction of gate partials
__global__ void reduce_gstats_kernel(const float* __restrict__ gpart,
                                     float* __restrict__ gstats) {
    const int c = threadIdx.x;
    if (c < 2) {
        float s = 0.f;
        for (int b = 0; b < TOTALN / 8; ++b) s += gpart[(size_t)b * 2 + c];
        gstats[c] = s;
    }
}

// ---------------------------------------------------------------------------
// K6: pass 3 — recompute h, BN+SiLU, softmax(gate) over K, weighted sum -> out
// ---------------------------------------------------------------------------
__global__ __launch_bounds__(256) void out_kernel(const float* __restrict__ x,
                                                  const int* __restrict__ nbr,
                                                  const float* __restrict__ W1,
                                                  const float* __restrict__ b1,
                                                  const float* __restrict__ g1,
                                                  const float* __restrict__ be1,
                                                  const float* __restrict__ hstats,
                                                  const float* __restrict__ gt_in,
                                                  const float* __restrict__ gstats,
                                                  const float* __restrict__ gg,
                                                  const float* __restrict__ beg,
                                                  float* __restrict__ out) {
    __shared__ float w1s[2 * CIN * OOUT];
    __shared__ float hb[8][16 * HPAD];
    __shared__ float attn[8][16];

    const int tid = threadIdx.x;
    stage_w1(W1, w1s, tid);
    __syncthreads();

    const int wave = tid >> 5, lane = tid & 31;
    const int node = blockIdx.x * 8 + wave;
    wave_compute_h(node, lane, x, nbr, w1s, b1, &hb[wave][0]);
    __syncthreads();
    apply_bn_silu(&hb[wave][0], lane, hstats, g1, be1);

    if (lane < 16) {
        const float invn = 1.0f / (float)NEDGE;
        const float gmu = gstats[0] * invn;
        const float gvar = gstats[1] * invn - gmu * gmu;
        const float grstd = rsqrtf(gvar + EPSBN);
        float v = gt_in[(size_t)node * KNN2 + lane];
        v = (v - gmu) * grstd * gg[0] + beg[0];
        v = v / (1.0f + __expf(-v));   // SiLU
        // softmax over the 16 edges (lanes 0..15)
        float mx = v;
#pragma unroll
        for (int mask = 1; mask < 16; mask <<= 1) mx = fmaxf(mx, __shfl_xor(mx, mask));
        float e = __expf(v - mx);
        float se = e;
#pragma unroll
        for (int mask = 1; mask < 16; mask <<= 1) se += __shfl_xor(se, mask);
        attn[wave][lane] = e / se;
    }
    __syncthreads();

#pragma unroll
    for (int half = 0; half < 2; ++half) {
        const int c = lane + 32 * half;
        float o = 0.f;
#pragma unroll
        for (int k = 0; k < 16; ++k) o += attn[wave][k] * hb[wave][k * HPAD + c];
        out[(size_t)node * OOUT + c] = o;
    }
}

// ---------------------------------------------------------------------------
// host launcher
// ---------------------------------------------------------------------------
extern "C" void kernel_launch(void* const* d_in, const int* in_sizes, int n_in,
                              void* d_out, int out_size, void* d_ws, size_t ws_size,
                              hipStream_t stream) {
    (void)in_sizes; (void)n_in; (void)out_size; (void)ws_size;
    const float* x   = (const float*)d_in[0];
    // d_in[1] = batch (sorted equal-size graphs -> implicit), unused
    const float* W1  = (const float*)d_in[2];
    const float* b1  = (const float*)d_in[3];
    const float* g1  = (const float*)d_in[4];
    const float* be1 = (const float*)d_in[5];
    const float* Wg  = (const float*)d_in[6];
    const float* bg  = (const float*)d_in[7];
    const float* gg  = (const float*)d_in[8];
    const float* beg = (const float*)d_in[9];
    float* out = (float*)d_out;

    float* wsf    = (float*)d_ws;
    float* hstats = wsf;                                   // 128 floats
    float* gstats = wsf + 128;                             // 2 floats (pad to 256)
    float* sq     = wsf + 256;                             // 32768
    int*   idx    = (int*)(wsf + 256 + TOTALN);            // 524288 ints
    float* gt     = wsf + 256 + TOTALN + NEDGE;            // 524288
    float* hpart  = wsf + 256 + TOTALN + 2 * NEDGE;        // 4096*128
    float* gpart  = hpart + (size_t)(TOTALN / 8) * 128;    // 8192

    sqnorm_kernel<<<TOTALN / 256, 256, 0, stream>>>(x, sq);
    knn_kernel<<<NGRAPH * (NNODE / 16), 128, 0, stream>>>(x, sq, idx);
    mlp_stats_kernel<<<TOTALN / 8, 256, 0, stream>>>(x, idx, W1, b1, hpart);
    reduce_hstats_kernel<<<1, 128, 0, stream>>>(hpart, hstats);
    gate_kernel<<<TOTALN / 8, 256, 0, stream>>>(x, idx, W1, b1, g1, be1, hstats,
                                                Wg, bg, gt, gpart);
    reduce_gstats_kernel<<<1, 64, 0, stream>>>(gpart, gstats);
    out_kernel<<<TOTALN / 8, 256, 0, stream>>>(x, idx, W1, b1, g1, be1, hstats,
                                               gt, gstats, gg, beg, out);
}